// LearnableMidiSynth_68771016343942
// MI455X (gfx1250) — compile-verified
//
#include <hip/hip_runtime.h>

typedef float v2f __attribute__((ext_vector_type(2)));
typedef float v8f __attribute__((ext_vector_type(8)));

__device__ __forceinline__ float fast_tanh(float x) {
#if __has_builtin(__builtin_amdgcn_tanhf)
    return __builtin_amdgcn_tanhf(x);          // gfx1250 v_tanh_f32 (confirmed in asm)
#elif __has_builtin(__builtin_amdgcn_tanh_f32)
    return __builtin_amdgcn_tanh_f32(x);
#else
    return tanhf(x);
#endif
}

// sin/cos at multiples {p, p+1, p+4, p+5} of `ang`, p = pf (odd: 1 or 3).
// 2 sincos + ~16 FMA via angle-addition / double-angle identities.
struct Quad { float sp, cp, sp1, cp1, sp4, cp4, sp5, cp5; };

__device__ __forceinline__ Quad harmonic_quad(float ang, float pf) {
    float s1, c1, sp, cp;
    sincosf(ang, &s1, &c1);            // sin/cos(ang)
    sincosf(pf * ang, &sp, &cp);       // sin/cos(p*ang)
    Quad q;
    q.sp  = sp;                q.cp  = cp;
    q.sp1 = sp * c1 + cp * s1; q.cp1 = cp * c1 - sp * s1;   // p+1
    const float s2 = 2.f * s1 * c1;
    const float c2 = 1.f - 2.f * s1 * s1;
    const float s4 = 2.f * s2 * c2;
    const float c4 = 1.f - 2.f * s2 * s2;                   // 4*ang
    q.sp4 = sp    * c4 + cp    * s4;  q.cp4 = cp    * c4 - sp    * s4;  // p+4
    q.sp5 = q.sp1 * c4 + q.cp1 * s4;  q.cp5 = q.cp1 * c4 - q.sp1 * s4;  // p+5
    return q;
}

__global__ void zero_f4_kernel(float4* __restrict__ p, int n4) {
    int i = blockIdx.x * blockDim.x + threadIdx.x;
    if (i < n4) p[i] = make_float4(0.f, 0.f, 0.f, 0.f);
}

// One block per note; wave w handles 256-sample tiles w, w+8, ...
// Tile factorization: t = t0 + 16m + n, seg = A(16x16) x B(16x16) done as
// four v_wmma_f32_16x16x4_f32 K-slices. Lane half hi only ever touches
// harmonic multiples {p,p+1,p+4,p+5}, p=2*hi+1 -> no selects, no arrays.
__global__ __launch_bounds__(256)
void synth_wmma_kernel(const float* __restrict__ freq_rad,
                       const float* __restrict__ amps,
                       const float* __restrict__ gain_p,
                       const int*   __restrict__ start_sample,
                       const int*   __restrict__ note_length,
                       float*       __restrict__ out,
                       int dur) {
    const int note = blockIdx.x;
    const int lane = threadIdx.x & 31;
    const int ln   = lane & 15;        // m-index (A side) == n-index (B side)
    const int hi   = lane >> 4;        // lane half selects K pair / D row offset
    const int wave = threadIdx.x >> 5;

    const float f     = freq_rad[note];
    const int   start = start_sample[note];
    const int   nlen  = note_length[note];
    if (start >= dur) return;                         // block-uniform
    const int out_len = min(start + nlen, dur) - start;
    if (out_len <= 0) return;                         // block-uniform

    const float gain = gain_p[0];
    const float pf   = (float)(2 * hi + 1);

    // amps this lane contributes to B (harmonics h = p-1, p, p+3, p+4)
    const float amp0 = amps[2 * hi + 0];
    const float amp1 = amps[2 * hi + 1];
    const float amp4 = amps[2 * hi + 4];
    const float amp5 = amps[2 * hi + 5];

    // ---- per-note B operands: Y_n = f*n ----
    const Quad qy = harmonic_quad(f * (float)ln, pf);
    v2f b0, b1, b2, b3;
    b0.x = amp0 * qy.cp;   b0.y = amp1 * qy.cp1;   // k<8  : amps*cos
    b1.x = amp4 * qy.cp4;  b1.y = amp5 * qy.cp5;
    b2.x = amp0 * qy.sp;   b2.y = amp1 * qy.sp1;   // k>=8 : amps*sin
    b3.x = amp4 * qy.sp4;  b3.y = amp5 * qy.sp5;

    // ---- X side: init at first tile, then rotate by Delta = f*2048 per tile ----
    Quad qx = harmonic_quad(f * (float)((wave << 8) + 16 * ln), pf);
    const Quad qd = harmonic_quad(f * 2048.f, pf);   // per-multiple step rotators

    for (int tile = wave; (tile << 8) < out_len; tile += 8) {
        const int t0 = tile << 8;

        v8f acc = {};
        v2f a;
        a.x = qx.sp;  a.y = qx.sp1;
        acc = __builtin_amdgcn_wmma_f32_16x16x4_f32(false, a, false, b0, (short)0, acc, false, false);
        a.x = qx.sp4; a.y = qx.sp5;
        acc = __builtin_amdgcn_wmma_f32_16x16x4_f32(false, a, false, b1, (short)0, acc, false, false);
        a.x = qx.cp;  a.y = qx.cp1;
        acc = __builtin_amdgcn_wmma_f32_16x16x4_f32(false, a, false, b2, (short)0, acc, false, false);
        a.x = qx.cp4; a.y = qx.cp5;
        acc = __builtin_amdgcn_wmma_f32_16x16x4_f32(false, a, false, b3, (short)0, acc, false, false);

        // C/D layout: VGPR j = row (j + 8*hi), column = ln.
        // t = t0 + 16j + 128*hi + ln  ->  one base pointer + 64B immediate offsets.
        float* const pbase = out + (start + t0 + 128 * hi + ln);
        if (t0 + 255 < out_len) {
            // fast path (all tiles except the note's last partial one): unguarded
#pragma unroll
            for (int j = 0; j < 8; ++j) {
                atomicAdd(pbase + 16 * j, fast_tanh(gain * acc[j]));
            }
        } else {
            const int tbase = t0 + 128 * hi + ln;
#pragma unroll
            for (int j = 0; j < 8; ++j) {
                if (tbase + 16 * j < out_len) {
                    atomicAdd(pbase + 16 * j, fast_tanh(gain * acc[j]));
                }
            }
        }

        // advance each multiple's phasor by its own step rotation
        Quad nx;
        nx.sp  = qx.sp  * qd.cp  + qx.cp  * qd.sp;   nx.cp  = qx.cp  * qd.cp  - qx.sp  * qd.sp;
        nx.sp1 = qx.sp1 * qd.cp1 + qx.cp1 * qd.sp1;  nx.cp1 = qx.cp1 * qd.cp1 - qx.sp1 * qd.sp1;
        nx.sp4 = qx.sp4 * qd.cp4 + qx.cp4 * qd.sp4;  nx.cp4 = qx.cp4 * qd.cp4 - qx.sp4 * qd.sp4;
        nx.sp5 = qx.sp5 * qd.cp5 + qx.cp5 * qd.sp5;  nx.cp5 = qx.cp5 * qd.cp5 - qx.sp5 * qd.sp5;
        qx = nx;
    }
}

extern "C" void kernel_launch(void* const* d_in, const int* in_sizes, int n_in,
                              void* d_out, int out_size, void* d_ws, size_t ws_size,
                              hipStream_t stream) {
    const float* freq_rad  = (const float*)d_in[0];
    const float* amps      = (const float*)d_in[1];
    const float* gain      = (const float*)d_in[2];
    const int*   start_smp = (const int*)d_in[3];
    const int*   note_len  = (const int*)d_in[4];
    float*       out       = (float*)d_out;

    const int N   = in_sizes[0];
    const int dur = out_size;

    const int n4 = dur >> 2;
    zero_f4_kernel<<<(n4 + 255) / 256, 256, 0, stream>>>((float4*)out, n4);

    synth_wmma_kernel<<<N, 256, 0, stream>>>(freq_rad, amps, gain,
                                             start_smp, note_len, out, dur);
}